// DCC_Loss_Function_81217831567711
// MI455X (gfx1250) — compile-verified
//
#include <hip/hip_runtime.h>
#include <hip/hip_bf16.h>
#include <stdint.h>

typedef __attribute__((ext_vector_type(2))) float v2f;
typedef __attribute__((ext_vector_type(8))) float v8f;

#define BB 64
#define CCH 3
#define HH 256
#define WW 256
#define NPIX (BB * HH * WW)           // 4194304
#define NELEM (BB * CCH * HH * WW)    // 12582912
#define C1F 0.0001f                   // 0.01^2
#define C2F 0.0009f                   // 0.03^2

// 11-tap Gaussian, sigma=1.5, normalized (matches reference _gauss_window)
__device__ __constant__ float G11[11] = {
    0.00102838f, 0.00759880f, 0.03600077f, 0.10936069f, 0.21300553f,
    0.26601172f,
    0.21300553f, 0.10936069f, 0.03600077f, 0.00759880f, 0.00102838f
};

// accumulator slots in d_ws (floats):
// 0 dehaze_abs  1 tv_h  2 tv_v  3 hist  4 cos_sum  5 mask_sum  6 ssim_sum

__device__ inline float block_sum(float v, float* sred) {
    const int tid = threadIdx.x;
    sred[tid] = v;
    __syncthreads();
    for (int s = 128; s > 0; s >>= 1) {
        if (tid < s) sred[tid] += sred[tid + s];
        __syncthreads();
    }
    float r = sred[0];
    __syncthreads();
    return r;
}

__global__ void k_init(float* acc) {
    if (threadIdx.x < 16) acc[threadIdx.x] = 0.f;
}

// ---------------- pointwise: L_dehaze, L_tv, L_normal terms ----------------
__global__ __launch_bounds__(256) void k_point(
    const float* __restrict__ predict, const float* __restrict__ gtruth,
    const float* __restrict__ dimg,    const float* __restrict__ cwat,
    const float* __restrict__ mask,    float* __restrict__ acc)
{
    __shared__ float sred[256];
    const int tid = threadIdx.x;
    const unsigned pix = blockIdx.x * 256u + tid;     // exactly NPIX threads
    const int b  = pix >> 16;
    const int hw = pix & 0xFFFF;
    const int h  = hw >> 8;
    const int w  = hw & 255;

    const size_t mi = ((size_t)b << 16) + hw;
    const float m  = mask[mi];
    const float mr = (w < 255) ? mask[mi + 1]   : 0.f;
    const float mb = (h < 255) ? mask[mi + 256] : 0.f;

    float s_de = 0.f, s_th = 0.f, s_tv = 0.f;
    float p[3], gt[3];
#pragma unroll
    for (int ch = 0; ch < 3; ++ch) {
        const size_t ii = ((size_t)(b * 3 + ch) << 16) + hw;
        const float d = dimg[ii] * m;
        const float c = cwat[ii] * m;
        s_de += fabsf(d - c);
        if (w < 255) s_th += fabsf(d - dimg[ii + 1]   * mr);
        if (h < 255) s_tv += fabsf(d - dimg[ii + 256] * mb);
        p[ch]  = predict[ii] * m;
        gt[ch] = gtruth[ii];
    }
    // normalize predict over channel dim
    const float pn = sqrtf(p[0]*p[0] + p[1]*p[1] + p[2]*p[2]);
    const float pd = fmaxf(pn, 1e-12f);
    const float ph0 = p[0] / pd, ph1 = p[1] / pd, ph2 = p[2] / pd;
    // gt normal: (gt*mask*2 - 1)*mask
    const float g0 = (gt[0]*m*2.f - 1.f) * m;
    const float g1 = (gt[1]*m*2.f - 1.f) * m;
    const float g2 = (gt[2]*m*2.f - 1.f) * m;
    const float dot = ph0*g0 + ph1*g1 + ph2*g2;
    const float n1 = fmaxf(sqrtf(ph0*ph0 + ph1*ph1 + ph2*ph2), 1e-8f);
    const float n2 = fmaxf(sqrtf(g0*g0 + g1*g1 + g2*g2), 1e-8f);
    const float cosv = 1.f - dot / (n1 * n2);

    float r;
    r = block_sum(s_de, sred); if (tid == 0) atomicAdd(&acc[0], r);
    r = block_sum(s_th, sred); if (tid == 0) atomicAdd(&acc[1], r);
    r = block_sum(s_tv, sred); if (tid == 0) atomicAdd(&acc[2], r);
    r = block_sum(cosv, sred); if (tid == 0) atomicAdd(&acc[4], r);
    r = block_sum(m,    sred); if (tid == 0) atomicAdd(&acc[5], r);
}

// ---------------- histogram: L_hist ----------------
__global__ __launch_bounds__(256) void k_hist(
    const float* __restrict__ gtruth, const float* __restrict__ nhist,
    float* __restrict__ acc)
{
    __shared__ unsigned hist[64];
    __shared__ float sred[256];
    const int tid = threadIdx.x;
    const int bc = blockIdx.x;                 // b*3 + c
    if (tid < 64) hist[tid] = 0u;
    __syncthreads();
    const size_t base = (size_t)bc << 16;
    for (int i = tid; i < 65536; i += 256) {
        if (i + 4096 < 65536)
            __builtin_prefetch(&gtruth[base + i + 4096], 0, 0);
        const float x = gtruth[base + i];
        int idx = (int)floorf(x * 64.f);       // == floor(((2x-1)+1)*0.5*64)
        idx = idx < 0 ? 0 : (idx > 63 ? 63 : idx);
        atomicAdd(&hist[idx], 1u);
    }
    __syncthreads();
    float dv = 0.f;
    if (tid < 64) {
        const float hv = (float)hist[tid] * (1.f / 65536.f);
        dv = fabsf(nhist[(size_t)bc * 64 + tid] - hv);
    }
    const float r = block_sum(dv, sred);
    if (tid == 0) atomicAdd(&acc[3], r);
}

// ---------------- SSIM via banded-matmul WMMA (separable Gaussian) --------
// One block per 64x64 output tile. Halo staged into LDS via the CDNA5
// async global->LDS engine (ASYNCcnt), then products formed in place.
// Horizontal pass: out = X(16x28) * Band(28x16) per 16x16 subtile;
// vertical pass: out = Band(16x28) * X(28x16). Both accumulate 7 chunks
// of K=4 on V_WMMA_F32_16X16X4_F32.
__global__ __launch_bounds__(256) void k_ssim(
    const float* __restrict__ dimg, const float* __restrict__ cwat,
    const float* __restrict__ mask, float* __restrict__ acc)
{
    extern __shared__ float smem[];
    float* mS  = smem;             // 80*80 raw mask tile
    float* dT  = mS + 6400;        // 80*80 raw dimg tile -> dehaze product
    float* cT  = dT + 6400;        // 80*80 raw cwat tile -> clean product
    float* Hf  = cT + 6400;        // 80*64 horizontal-pass intermediate
    float* gp  = Hf + 5120;        // 48    padded band taps, gp[16+i]=g[i]
    float* red = gp + 48;          // 256

    const int tid  = threadIdx.x;
    const int wave = tid >> 5;
    const int lane = tid & 31;
    const int hi   = lane >> 4;    // lane-half select per WMMA layouts
    const int m16  = lane & 15;

    const int img  = blockIdx.x >> 4;          // b*3 + ch
    const int tile = blockIdx.x & 15;
    const int b    = img / 3;
    const int r0   = (tile >> 2) * 64;
    const int c0   = (tile & 3) * 64;

    if (tid < 48) {
        const int i = tid - 16;
        gp[tid] = (i >= 0 && i < 11) ? G11[i] : 0.f;
    }

    // ---- stage halo via async global->LDS (zero padding kept exact) ----
    const size_t ibase = (size_t)img << 16;
    const size_t mbase = (size_t)b << 16;
    for (int idx = tid; idx < 6400; idx += 256) {
        mS[idx] = 0.f; dT[idx] = 0.f; cT[idx] = 0.f;
    }
    __syncthreads();
    asm volatile("s_wait_dscnt 0x0" ::: "memory");
    for (int idx = tid; idx < 6400; idx += 256) {
        const int r  = idx / 80;
        const int cc = idx - r * 80;
        const int gr = r0 - 5 + r;
        const int gc = c0 - 5 + cc;
        if (r < 74 && cc < 74 && gr >= 0 && gr < 256 && gc >= 0 && gc < 256) {
            const size_t off = ((size_t)gr << 8) + gc;
            const unsigned long long ga_m =
                (unsigned long long)(uintptr_t)(mask + mbase + off);
            const unsigned long long ga_d =
                (unsigned long long)(uintptr_t)(dimg + ibase + off);
            const unsigned long long ga_c =
                (unsigned long long)(uintptr_t)(cwat + ibase + off);
            const unsigned lm = (unsigned)(uintptr_t)&mS[idx];
            const unsigned ld = (unsigned)(uintptr_t)&dT[idx];
            const unsigned lc = (unsigned)(uintptr_t)&cT[idx];
            asm volatile("global_load_async_to_lds_b32 %0, %1, off"
                         :: "v"(lm), "v"(ga_m) : "memory");
            asm volatile("global_load_async_to_lds_b32 %0, %1, off"
                         :: "v"(ld), "v"(ga_d) : "memory");
            asm volatile("global_load_async_to_lds_b32 %0, %1, off"
                         :: "v"(lc), "v"(ga_c) : "memory");
        }
    }
    asm volatile("s_wait_asynccnt 0x0" ::: "memory");
    __syncthreads();
    // products in place: dehaze = dimg*mask, clean = cwat*mask
    for (int idx = tid; idx < 6400; idx += 256) {
        const float m = mS[idx];
        dT[idx] *= m;
        cT[idx] *= m;
    }

    v8f res0[5], res1[5];
#pragma unroll
    for (int f = 0; f < 5; ++f) {
        __syncthreads();
        // horizontal pass: 20 subtiles (5 row-tiles of 16 x 4 col-tiles)
        for (int task = wave; task < 20; task += 8) {
            const int rt = (task >> 2) << 4;   // 0..64
            const int ct = (task & 3) << 4;    // 0..48
            v8f accv = {};
#pragma unroll
            for (int s = 0; s < 7; ++s) {
                const int kk0 = 4 * s + 2 * hi;
                v2f a, bb;
#pragma unroll
                for (int v = 0; v < 2; ++v) {
                    const int col = ct + kk0 + v;          // halo col, <=75
                    const float dv = dT[(rt + m16) * 80 + col];
                    const float cv = cT[(rt + m16) * 80 + col];
                    float fv;
                    if      (f == 0) fv = dv;
                    else if (f == 1) fv = cv;
                    else if (f == 2) fv = dv * dv;
                    else if (f == 3) fv = cv * cv;
                    else             fv = dv * cv;
                    a[v]  = fv;                            // A[M][2hi+v]
                    bb[v] = gp[16 + (kk0 + v) - m16];      // Band[k][N]
                }
                accv = __builtin_amdgcn_wmma_f32_16x16x4_f32(
                    false, a, false, bb, (short)0, accv, false, false);
            }
#pragma unroll
            for (int v = 0; v < 8; ++v)
                Hf[(rt + v + 8 * hi) * 64 + ct + m16] = accv[v];
        }
        __syncthreads();
        // vertical pass: 16 output subtiles, 2 per wave
#pragma unroll
        for (int t = 0; t < 2; ++t) {
            const int st = wave * 2 + t;
            const int it = (st >> 2) << 4;
            const int jt = (st & 3) << 4;
            v8f accv = {};
#pragma unroll
            for (int s = 0; s < 7; ++s) {
                const int kk0 = 4 * s + 2 * hi;
                v2f a, bb;
#pragma unroll
                for (int v = 0; v < 2; ++v) {
                    a[v]  = gp[16 + (kk0 + v) - m16];      // Band[M][k]
                    bb[v] = Hf[(it + kk0 + v) * 64 + jt + m16];
                }
                accv = __builtin_amdgcn_wmma_f32_16x16x4_f32(
                    false, a, false, bb, (short)0, accv, false, false);
            }
            if (t == 0) res0[f] = accv; else res1[f] = accv;
        }
    }

    // combine SSIM map and reduce
    auto combine = [&](const v8f* R) {
        float s = 0.f;
#pragma unroll
        for (int v = 0; v < 8; ++v) {
            const float mu1 = R[0][v], mu2 = R[1][v];
            const float e11 = R[2][v], e22 = R[3][v], e12 = R[4][v];
            const float m11 = mu1 * mu1, m22 = mu2 * mu2, m12 = mu1 * mu2;
            const float num = (2.f * m12 + C1F) * (2.f * (e12 - m12) + C2F);
            const float den = (m11 + m22 + C1F) *
                              ((e11 - m11) + (e22 - m22) + C2F);
            s += num / den;
        }
        return s;
    };
    const float ssum = combine(res0) + combine(res1);
    const float r = block_sum(ssum, red);
    if (tid == 0) atomicAdd(&acc[6], r);
}

// ---------------- final scalar ----------------
__global__ void k_final(const float* __restrict__ acc, float* __restrict__ out) {
    if (threadIdx.x == 0 && blockIdx.x == 0) {
        const float Ninv = 1.f / (float)NELEM;
        const float L_de   = acc[0] * Ninv;
        const float L_ssim = 1.f - acc[6] * Ninv;
        const float tvden  = 1.f / (float)(BB * CCH * HH * (WW - 1));
        const float L_tv   = acc[1] * tvden + acc[2] * tvden;
        const float L_hist = acc[3] * (1.f / (float)(BB * CCH * 64));
        const float M      = acc[5];
        const float back   = (float)(BB * HH * WW) - M;
        const float L_norm = (acc[4] - back) / M;
        out[0] = 10.f * L_de + L_ssim + L_tv + L_hist + 100.f * L_norm;
    }
}

extern "C" void kernel_launch(void* const* d_in, const int* in_sizes, int n_in,
                              void* d_out, int out_size, void* d_ws, size_t ws_size,
                              hipStream_t stream) {
    const float* predict = (const float*)d_in[0];
    const float* gtruth  = (const float*)d_in[1];
    const float* dimg    = (const float*)d_in[2];
    const float* cwat    = (const float*)d_in[3];
    const float* nhist   = (const float*)d_in[4];
    const float* mask    = (const float*)d_in[5];
    float* acc = (float*)d_ws;
    float* out = (float*)d_out;

    hipLaunchKernelGGL(k_init, dim3(1), dim3(32), 0, stream, acc);
    hipLaunchKernelGGL(k_point, dim3(NPIX / 256), dim3(256), 0, stream,
                       predict, gtruth, dimg, cwat, mask, acc);
    hipLaunchKernelGGL(k_hist, dim3(BB * CCH), dim3(256), 0, stream,
                       gtruth, nhist, acc);
    // dyn LDS: (6400*3+5120+48+256)*4 = 98496 bytes
    hipLaunchKernelGGL(k_ssim, dim3(BB * CCH * 16), dim3(256), 98496, stream,
                       dimg, cwat, mask, acc);
    hipLaunchKernelGGL(k_final, dim3(1), dim3(32), 0, stream, acc, out);
}